// DFineMultiscaleDeformableAttention_64544768524462
// MI455X (gfx1250) — compile-verified
//
#include <hip/hip_runtime.h>
#include <hip/hip_bf16.h>

// ---------------- problem constants (from reference) ----------------
#define BQ      4800        // B*Q = 16*300
#define DM      256         // D_MODEL
#define NH      8           // heads
#define HD      32          // head dim
#define SP      16          // sample points per head
#define NOFF    256         // TP*2
#define NATT    128         // TP
#define NT_TOT  24          // 16 off n-tiles + 8 att n-tiles
#define KSTEPS  64          // 256 / 4
#define S_TOT   13294
#define Q_      300

#define OUT_ATT_OFF (BQ * DM)           // attw starts here in d_out (floats)
#define WS_OFF_OFF  0
#define WS_LOG_OFF  (BQ * NOFF)         // 1228800
#define WS_PACK_OFF (BQ * NOFF + BQ * NATT)  // 1843200
#define PACK_ELEMS  (NT_TOT * KSTEPS * 64)   // 98304

typedef float v2f __attribute__((ext_vector_type(2)));
typedef float v8f __attribute__((ext_vector_type(8)));

__device__ __forceinline__ int iclamp(int v, int lo, int hi) {
    return v < lo ? lo : (v > hi ? hi : v);
}

// -------------------------------------------------------------------
// Kernel 1: repack [w_off | w_att] into V_WMMA_F32_16X16X4_F32 B-fragment
// layout: pack[t][ks][lane][e], lane<16 holds B[k..k+1][n], lane>=16 holds
// B[k+2..k+3][n], n = lane&15, k = 4*ks.
// -------------------------------------------------------------------
__global__ __launch_bounds__(256) void repack_w_kernel(
    const float* __restrict__ w_off, const float* __restrict__ w_att,
    float* __restrict__ pack)
{
    int p = blockIdx.x * 256 + threadIdx.x;
    if (p >= PACK_ELEMS) return;
    int e    = p & 1;
    int lane = (p >> 1) & 31;
    int ks   = (p >> 6) & (KSTEPS - 1);
    int t    = p >> 12;
    int n    = lane & 15;
    int k    = ks * 4 + ((lane >> 4) << 1) + e;
    float v;
    if (t < 16) v = w_off[(size_t)k * NOFF + t * 16 + n];
    else        v = w_att[(size_t)k * NATT + (t - 16) * 16 + n];
    pack[p] = v;
}

// -------------------------------------------------------------------
// Kernel 2: fused GEMM  P = hidden @ [w_off | w_att] + bias  via fp32 WMMA.
// Block = 256 threads (8 waves) handles one 16-row M tile; A staged in LDS
// (row stride 260 -> conflict-free ds_load_b64 fragments); each wave does
// 3 of the 24 N tiles.  D tile layout: VGPR r, lanes 0-15 -> row r,
// lanes 16-31 -> row r+8, col = lane&15.
// -------------------------------------------------------------------
__global__ __launch_bounds__(256) void gemm_wmma_kernel(
    const float* __restrict__ hid, const float* __restrict__ pack,
    const float* __restrict__ b_off, const float* __restrict__ b_att,
    float* __restrict__ out_off, float* __restrict__ out_log)
{
    __shared__ float As[16 * 260];
    const int m0 = blockIdx.x * 16;

    // cooperative load of A tile (16 x 256)
    for (int i = threadIdx.x; i < 16 * 256; i += 256) {
        int r = i >> 8, c = i & 255;
        As[r * 260 + c] = hid[(size_t)(m0 + r) * DM + c];
    }
    __syncthreads();

    const int wave = threadIdx.x >> 5;
    const int lane = threadIdx.x & 31;
    const int arow = lane & 15;          // A fragment row
    const int koff = (lane >> 4) << 1;   // 0 or 2
    const int ncol = lane & 15;          // output column within tile

    for (int t = wave; t < NT_TOT; t += 8) {
        float bias = (t < 16) ? b_off[t * 16 + ncol]
                              : b_att[(t - 16) * 16 + ncol];
        v8f acc;
#pragma unroll
        for (int r = 0; r < 8; ++r) acc[r] = bias;

        const float* pB = pack + ((size_t)t * KSTEPS) * 64 + lane * 2;
#pragma unroll 8
        for (int ks = 0; ks < KSTEPS; ++ks) {
            v2f a = *(const v2f*)&As[arow * 260 + ks * 4 + koff];
            v2f b = *(const v2f*)&pB[(size_t)ks * 64];
            acc = __builtin_amdgcn_wmma_f32_16x16x4_f32(
                false, a, false, b, (short)0, acc, false, false);
        }

        const int mrow = m0 + ((lane < 16) ? 0 : 8);
        if (t < 16) {
            float* dst = out_off + (size_t)mrow * NOFF + t * 16 + ncol;
#pragma unroll
            for (int r = 0; r < 8; ++r) dst[(size_t)r * NOFF] = acc[r];
        } else {
            float* dst = out_log + (size_t)mrow * NATT + (t - 16) * 16 + ncol;
#pragma unroll
            for (int r = 0; r < 8; ++r) dst[(size_t)r * NATT] = acc[r];
        }
    }
}

// -------------------------------------------------------------------
// Kernel 3: softmax over SP=16 logits per (b,q,h); writes attw output.
// -------------------------------------------------------------------
__global__ __launch_bounds__(256) void softmax_kernel(
    const float* __restrict__ logits, float* __restrict__ attw)
{
    int i = blockIdx.x * 256 + threadIdx.x;   // (b*Q+q)*H + h
    if (i >= BQ * NH) return;
    const float* src = logits + (size_t)i * SP;
    float v[SP];
    float mx = -__builtin_huge_valf();
#pragma unroll
    for (int p = 0; p < SP; ++p) { v[p] = src[p]; mx = fmaxf(mx, v[p]); }
    float sum = 0.f;
#pragma unroll
    for (int p = 0; p < SP; ++p) { v[p] = __expf(v[p] - mx); sum += v[p]; }
    float inv = 1.f / sum;
    float* dst = attw + (size_t)i * SP;
#pragma unroll
    for (int p = 0; p < SP; ++p) dst[p] = v[p] * inv;
}

// -------------------------------------------------------------------
// Kernel 4: deformable bilinear sampling + weighted reduction.
// One wave per (b,q,h); lane = channel d (HD=32).  Every corner load is a
// coalesced 128B transaction from the L2-resident encoder states.
// -------------------------------------------------------------------
__global__ __launch_bounds__(256) void sample_kernel(
    const float* __restrict__ enc, const float* __restrict__ refp,
    const float* __restrict__ off, const float* __restrict__ attw,
    float* __restrict__ out)
{
    const int wid  = blockIdx.x * 8 + (threadIdx.x >> 5);   // 0 .. BQ*NH-1
    const int lane = threadIdx.x & 31;
    const int h    = wid & 7;
    const int bq   = wid >> 3;
    const int b    = bq / Q_;

    const float4 ref = *(const float4*)(refp + (size_t)bq * 4);
    const float* offp  = off  + (size_t)bq * NOFF + h * 2 * SP;  // (h*16+p)*2
    const float* awp   = attw + (size_t)bq * NATT + h * SP;
    const float* vbase = enc + (size_t)b * S_TOT * DM + h * HD + lane;

    const int   LW[4] = {100, 50, 25, 13};
    const int   LH[4] = {100, 50, 25, 13};
    const int   LS[4] = {0, 10000, 12500, 13125};

    float acc = 0.f;
#pragma unroll
    for (int p = 0; p < SP; ++p) {
        const int lvl = p >> 2;
        const int W = LW[lvl], Hh = LH[lvl], st = LS[lvl];
        const float ox = offp[2 * p], oy = offp[2 * p + 1];
        const float aw = awp[p];
        // nps(=1/4) * OFFSET_SCALE(=0.5) = 0.125
        const float lx = ref.x + ox * 0.125f * ref.z;
        const float ly = ref.y + oy * 0.125f * ref.w;
        // grid_sample align_corners=False:  x = ((g+1)*W - 1)/2, g = 2*loc-1
        const float x = lx * (float)W - 0.5f;
        const float y = ly * (float)Hh - 0.5f;
        const float x0f = floorf(x), y0f = floorf(y);
        const float wx = x - x0f, wy = y - y0f;
        const int x0 = (int)x0f, y0 = (int)y0f;

        float s = 0.f;
#pragma unroll
        for (int c = 0; c < 4; ++c) {
            const int dx = c & 1, dy = c >> 1;
            const int xi = x0 + dx, yi = y0 + dy;
            const float wgt = (dx ? wx : 1.f - wx) * (dy ? wy : 1.f - wy);
            const bool valid = (xi >= 0) & (xi < W) & (yi >= 0) & (yi < Hh);
            const int sidx = st + iclamp(yi, 0, Hh - 1) * W + iclamp(xi, 0, W - 1);
            const float v = vbase[(size_t)sidx * DM];
            s += valid ? wgt * v : 0.f;
        }
        acc += aw * s;
    }
    out[(size_t)bq * DM + h * HD + lane] = acc;
}

// -------------------------------------------------------------------
extern "C" void kernel_launch(void* const* d_in, const int* in_sizes, int n_in,
                              void* d_out, int out_size, void* d_ws, size_t ws_size,
                              hipStream_t stream) {
    const float* hid   = (const float*)d_in[0];
    const float* enc   = (const float*)d_in[1];
    const float* refp  = (const float*)d_in[2];
    const float* w_off = (const float*)d_in[3];
    const float* b_off = (const float*)d_in[4];
    const float* w_att = (const float*)d_in[5];
    const float* b_att = (const float*)d_in[6];

    float* out  = (float*)d_out;                 // [B,Q,256]
    float* attw = out + OUT_ATT_OFF;             // [B,Q,8,16]

    float* ws      = (float*)d_ws;
    float* ws_off  = ws + WS_OFF_OFF;            // [BQ,256] raw offsets
    float* ws_log  = ws + WS_LOG_OFF;            // [BQ,128] attention logits
    float* ws_pack = ws + WS_PACK_OFF;           // packed WMMA B fragments

    repack_w_kernel<<<(PACK_ELEMS + 255) / 256, 256, 0, stream>>>(w_off, w_att, ws_pack);
    gemm_wmma_kernel<<<BQ / 16, 256, 0, stream>>>(hid, ws_pack, b_off, b_att,
                                                  ws_off, ws_log);
    softmax_kernel<<<(BQ * NH + 255) / 256, 256, 0, stream>>>(ws_log, attw);
    sample_kernel<<<(BQ * NH) / 8, 256, 0, stream>>>(enc, refp, ws_off, attw, out);
}